// RetinaEncoder_26405458936296
// MI455X (gfx1250) — compile-verified
//
#include <hip/hip_runtime.h>
#include <hip/hip_bf16.h>
#include <math.h>

#define MAX_GT     1024
#define NEG_THRESH 0.4f
#define POS_THRESH 0.5f

typedef __attribute__((ext_vector_type(4))) unsigned int u32x4;
typedef __attribute__((ext_vector_type(8))) int          i32x8;
typedef __attribute__((ext_vector_type(4))) int          i32x4;

#if defined(__has_builtin)
#  if __has_builtin(__builtin_amdgcn_tensor_load_to_lds) && __has_builtin(__builtin_amdgcn_s_wait_tensorcnt)
#    define USE_TDM 1
#  endif
#endif
#ifndef USE_TDM
#  define USE_TDM 0
#endif

// ---------------------------------------------------------------------------
// Kernel 1: per-prior max/argmax IoU over all GT boxes (argmax over axis=0).
// GT boxes staged into LDS via the CDNA5 Tensor Data Mover (TDM).
// ---------------------------------------------------------------------------
__global__ __launch_bounds__(256) void colmax_kernel(
    const float* __restrict__ bboxes,   // [N,4] xyxy
    const float* __restrict__ priors,   // [M,4] cxcywh
    float* __restrict__ max_ious,       // [M]
    int*   __restrict__ match_ids,      // [M]
    int*   __restrict__ override_n,     // [M] (init to -1 here)
    int M, int N)
{
    __shared__ float4 sbox[MAX_GT];     // 16 KB: GT boxes, xyxy
    __shared__ float  sarea[MAX_GT];    //  4 KB: GT areas
    const int tid = threadIdx.x;
    const int n4  = N * 4;

#if USE_TDM
    if (tid < 32) {                         // wave 0 only: issue one DMA
        unsigned lds_off = (unsigned)(unsigned long long)&sbox[0];
        unsigned long long ga = (unsigned long long)bboxes;
        u32x4 g0;
        g0.x = 1u;                                          // count=1 (valid user D#)
        g0.y = lds_off;                                     // lds_addr
        g0.z = (unsigned)(ga & 0xFFFFFFFFull);              // global_addr[31:0]
        g0.w = (unsigned)((ga >> 32) & 0x01FFFFFFull)       // global_addr[56:32]
             | 0x80000000u;                                 // type=2 ("image")
        i32x8 g1;
        g1[0] = 0x00020000;                                 // data_size=4B; mask/pad/iter=0
        g1[1] = (int)(((unsigned)n4 & 0xFFFFu) << 16);      // tensor_dim0[15:0]
        g1[2] = (int)((((unsigned)n4 >> 16) & 0xFFFFu)      // tensor_dim0[31:16]
             | (1u << 16));                                 // tensor_dim1 = 1
        g1[3] = (int)(((unsigned)n4 & 0xFFFFu) << 16);      // tile_dim0 = n4
        g1[4] = 0;                                          // tile_dim1/2 unused
        g1[5] = n4;                                         // tensor_dim0_stride lo
        g1[6] = 0;
        g1[7] = 0;
        i32x4 z4 = {0, 0, 0, 0};
#if __clang_major__ >= 23
        i32x8 z8 = {0, 0, 0, 0, 0, 0, 0, 0};
        __builtin_amdgcn_tensor_load_to_lds(g0, g1, z4, z4, z8, 0);
#else
        __builtin_amdgcn_tensor_load_to_lds(g0, g1, z4, z4, 0);
#endif
        __builtin_amdgcn_s_wait_tensorcnt(0);               // wave-private TENSORcnt
    }
    __syncthreads();                                        // publish LDS to all waves
#else
    {   // fallback: cooperative vector load
        float* dst = (float*)&sbox[0];
        for (int i = tid; i < n4; i += 256) dst[i] = bboxes[i];
        __syncthreads();
    }
#endif

    for (int n = tid; n < N; n += 256) {
        float4 b = sbox[n];
        sarea[n] = (b.z - b.x) * (b.w - b.y);
    }
    __syncthreads();

    const int m = blockIdx.x * 256 + tid;
    if (m >= M) return;

    const float4 p = ((const float4*)priors)[m];
    const float px1 = p.x - 0.5f * p.z, py1 = p.y - 0.5f * p.w;
    const float px2 = p.x + 0.5f * p.z, py2 = p.y + 0.5f * p.w;
    const float parea = p.z * p.w;

    // Track best as (inter, union) pair; compare by cross-multiplication to
    // avoid a v_rcp per element. Strict '>' keeps the FIRST max (jnp.argmax).
    float bi = -1.0f, bu = 1.0f;
    int   bn = 0;
#pragma unroll 4
    for (int n = 0; n < N; ++n) {
        float4 g = sbox[n];                 // broadcast ds_load_b128
        float x1 = fmaxf(px1, g.x), y1 = fmaxf(py1, g.y);
        float x2 = fminf(px2, g.z), y2 = fminf(py2, g.w);
        float w  = fmaxf(x2 - x1, 0.0f),  h = fmaxf(y2 - y1, 0.0f);
        float inter = w * h;
        float uni   = parea + sarea[n] - inter;
        bool better = inter * bu > bi * uni;
        bi = better ? inter : bi;
        bu = better ? uni   : bu;
        bn = better ? n     : bn;
    }
    max_ious[m]   = bi / bu;    // single division per prior
    match_ids[m]  = bn;
    override_n[m] = -1;
}

// ---------------------------------------------------------------------------
// Kernel 2: per-GT argmax IoU over all priors (argmax over axis=1).
// One block per GT; priors stream through L2 (entire array is L2-resident).
// ---------------------------------------------------------------------------
__global__ __launch_bounds__(256) void rowmax_kernel(
    const float* __restrict__ bboxes,
    const float* __restrict__ priors,
    int* __restrict__ force_ids,        // [N]
    int M)
{
    __shared__ float s_i[256], s_u[256];
    __shared__ int   s_n[256];
    const int n = blockIdx.x;
    const float gx1 = bboxes[n * 4 + 0], gy1 = bboxes[n * 4 + 1];
    const float gx2 = bboxes[n * 4 + 2], gy2 = bboxes[n * 4 + 3];
    const float garea = (gx2 - gx1) * (gy2 - gy1);

    const float4* p4 = (const float4*)priors;
    float bi = -1.0f, bu = 1.0f;
    int   bm = 0;
    for (int m = threadIdx.x; m < M; m += 256) {
        __builtin_prefetch((const void*)(p4 + m + 2048), 0, 1); // global_prefetch_b8
        float4 p = p4[m];
        float px1 = p.x - 0.5f * p.z, py1 = p.y - 0.5f * p.w;
        float px2 = p.x + 0.5f * p.z, py2 = p.y + 0.5f * p.w;
        float x1 = fmaxf(px1, gx1), y1 = fmaxf(py1, gy1);
        float x2 = fminf(px2, gx2), y2 = fminf(py2, gy2);
        float w  = fmaxf(x2 - x1, 0.0f), h = fmaxf(y2 - y1, 0.0f);
        float inter = w * h;
        float uni   = garea + p.z * p.w - inter;
        bool better = inter * bu > bi * uni;   // ascending m: ties keep lowest
        if (better) { bi = inter; bu = uni; bm = m; }
    }
    s_i[threadIdx.x] = bi; s_u[threadIdx.x] = bu; s_n[threadIdx.x] = bm;
    __syncthreads();
    for (int off = 128; off > 0; off >>= 1) {
        if (threadIdx.x < off) {
            float oi = s_i[threadIdx.x + off], ou = s_u[threadIdx.x + off];
            int   om = s_n[threadIdx.x + off];
            float a = oi * s_u[threadIdx.x];
            float b = s_i[threadIdx.x] * ou;
            if (a > b || (a == b && om < s_n[threadIdx.x])) {
                s_i[threadIdx.x] = oi; s_u[threadIdx.x] = ou; s_n[threadIdx.x] = om;
            }
        }
        __syncthreads();
    }
    if (threadIdx.x == 0) force_ids[n] = s_n[0];
}

// ---------------------------------------------------------------------------
// Kernel 3: force-assign scatter (highest GT index wins on duplicates).
// ---------------------------------------------------------------------------
__global__ void scatter_kernel(const int* __restrict__ force_ids,
                               int* __restrict__ override_n, int N)
{
    int n = blockIdx.x * blockDim.x + threadIdx.x;
    if (n < N) atomicMax(&override_n[force_ids[n]], n);
}

// ---------------------------------------------------------------------------
// Kernel 4: delta encoding + class targets.
// ---------------------------------------------------------------------------
__global__ __launch_bounds__(256) void encode_kernel(
    const float* __restrict__ bboxes,
    const int*   __restrict__ labels,
    const float* __restrict__ priors,
    const float* __restrict__ max_ious,
    const int*   __restrict__ match_ids,
    const int*   __restrict__ override_n,
    float* __restrict__ out, int M)
{
    const int m = blockIdx.x * 256 + threadIdx.x;
    if (m >= M) return;

    const int   ov   = override_n[m];
    const int   mid  = (ov >= 0) ? ov : match_ids[m];
    const float miou = (ov >= 0) ? POS_THRESH : max_ious[m];

    const float x1 = bboxes[mid * 4 + 0], y1 = bboxes[mid * 4 + 1];
    const float x2 = bboxes[mid * 4 + 2], y2 = bboxes[mid * 4 + 3];
    const float mcx = 0.5f * (x1 + x2), mcy = 0.5f * (y1 + y2);
    const float mw  = x2 - x1,          mh  = y2 - y1;

    const float4 p = ((const float4*)priors)[m];
    out[m * 4 + 0] = (mcx - p.x) / p.z * 10.0f;   // /0.1
    out[m * 4 + 1] = (mcy - p.y) / p.w * 10.0f;
    out[m * 4 + 2] = __logf(mw / p.z) * 5.0f;     // /0.2
    out[m * 4 + 3] = __logf(mh / p.w) * 5.0f;

    float cls = (float)labels[mid];
    if (miou < POS_THRESH) cls = -1.0f;
    if (miou < NEG_THRESH) cls = 0.0f;
    out[4 * M + m] = cls;
}

extern "C" void kernel_launch(void* const* d_in, const int* in_sizes, int n_in,
                              void* d_out, int out_size, void* d_ws, size_t ws_size,
                              hipStream_t stream) {
    const float* bboxes = (const float*)d_in[0];   // [N,4] f32 xyxy
    const int*   labels = (const int*)  d_in[1];   // [N]   i32
    const float* priors = (const float*)d_in[2];   // [M,4] f32 cxcywh
    const int N = in_sizes[0] / 4;
    const int M = in_sizes[2] / 4;

    // Workspace layout: max_ious[M] f32 | match_ids[M] i32 | override[M] i32 | force[N] i32
    float* ws_max   = (float*)d_ws;
    int*   ws_match = (int*)(ws_max + M);
    int*   ws_over  = ws_match + M;
    int*   ws_force = ws_over + M;
    float* out      = (float*)d_out;

    dim3 blk(256);
    colmax_kernel <<<(M + 255) / 256, blk, 0, stream>>>(bboxes, priors, ws_max, ws_match, ws_over, M, N);
    rowmax_kernel <<<N,               blk, 0, stream>>>(bboxes, priors, ws_force, M);
    scatter_kernel<<<(N + 255) / 256, blk, 0, stream>>>(ws_force, ws_over, N);
    encode_kernel <<<(M + 255) / 256, blk, 0, stream>>>(bboxes, labels, priors,
                                                        ws_max, ws_match, ws_over, out, M);
}